// DEQSolver_2894807957574
// MI455X (gfx1250) — compile-verified
//
#include <hip/hip_runtime.h>

#define BB 8
#define DD ((size_t)(3 * 512 * 512))   // 786432 elements per batch (multiple of 256)
#define MH 5                           // Anderson history m
#define FTHRES 40
#define ANDLAM 1e-4f
#define GBLK 16                        // gram blocks per batch (x 8 waves = 128 waves/batch)

typedef __attribute__((ext_vector_type(2))) float v2f;
typedef __attribute__((ext_vector_type(8))) float v8f;

__device__ __forceinline__ float shrinkf(float u, float t) {
    float a = fabsf(u) - t;
    a = a > 0.f ? a : 0.f;
    return copysignf(a, u);   // sign(u)*max(|u|-t,0); u==0 -> 0 anyway
}

// F[slot] = ista(zin); G[slot] = F[slot] - zin   (zin batch stride parametric)
// grid = dim3(DD/256, BB), block = 256
__global__ void fstep_kernel(const float* __restrict__ zin, size_t zstride,
                             float* __restrict__ F, float* __restrict__ G, int slot,
                             const float* __restrict__ x0,
                             const float* __restrict__ rho_p, const float* __restrict__ lam_p) {
    size_t e = (size_t)blockIdx.x * blockDim.x + threadIdx.x;
    size_t b = blockIdx.y;
    float rho = *rho_p, lam = *lam_p;
    float z = zin[b * zstride + e];
    float x = x0[b * DD + e];
    float u = z - rho * (z - x);
    float zn = shrinkf(u, rho * lam);
    size_t so = (b * MH + (size_t)slot) * DD + e;
    F[so] = zn;
    G[so] = zn - z;
}

__global__ void zero_kernel(float* __restrict__ p, int nElem) {
    int i = blockIdx.x * blockDim.x + threadIdx.x;
    if (i < nElem) p[i] = 0.f;
}

// Per-batch Gram of G rows 0..n-1 via V_WMMA_F32_16X16X4_F32.
// A = G (16x4, rows >= n fed duplicate row-0 data: pollutes only unused D rows/cols),
// B = G^T (4x16): per-lane registers identical to A, so one load feeds both.
// grid = dim3(GBLK, BB), block = 256 (8 waves)
__global__ void gram_kernel(const float* __restrict__ G, float* __restrict__ gram, int n) {
    int b    = blockIdx.y;
    int lane = threadIdx.x & 31;
    int wave = blockIdx.x * (blockDim.x >> 5) + (threadIdx.x >> 5);
    const int wavesPerBatch = GBLK * 8;           // 128
    const size_t span = DD / wavesPerBatch;       // 6144, multiple of 16
    size_t k0 = (size_t)wave * span;

    int row = lane & 15;                 // A-row / B-col this lane feeds
    int kh  = (lane >> 4) << 1;          // lanes 0-15: K=0,1 ; lanes 16-31: K=2,3
    int rr  = (row < n) ? row : 0;       // clamp: junk rows touch only unread D entries
    const float* p = G + ((size_t)b * MH + rr) * DD + k0 + (size_t)kh;

    v8f acc = {};
    for (size_t kk = 0; kk < span; kk += 16) {
        v2f a0 = *(const v2f*)(p + kk);
        v2f a1 = *(const v2f*)(p + kk + 4);
        v2f a2 = *(const v2f*)(p + kk + 8);
        v2f a3 = *(const v2f*)(p + kk + 12);
        acc = __builtin_amdgcn_wmma_f32_16x16x4_f32(false, a0, false, a0, (short)0, acc, false, false);
        acc = __builtin_amdgcn_wmma_f32_16x16x4_f32(false, a1, false, a1, (short)0, acc, false, false);
        acc = __builtin_amdgcn_wmma_f32_16x16x4_f32(false, a2, false, a2, (short)0, acc, false, false);
        acc = __builtin_amdgcn_wmma_f32_16x16x4_f32(false, a3, false, a3, (short)0, acc, false, false);
    }

    // D layout: lane<16 -> VGPR v holds D[M=v][N=lane]; lane>=16 -> D[M=v+8][N=lane-16]
    int col = lane & 15;
    int rb  = (lane >> 4) << 3;
    if (col < n) {
        #pragma unroll
        for (int v = 0; v < 8; ++v) {
            int r = rb + v;
            if (r < n)
                atomicAdd(&gram[((size_t)b * MH + r) * MH + col], acc[v]);
        }
    }
}

// Bordered (n+1)x(n+1) solve per batch, one lane per batch.
__global__ void solve_kernel(const float* __restrict__ gram, float* __restrict__ alpha, int n) {
    int b = threadIdx.x;
    if (b >= BB) return;
    float A[6][7];
    int N = n + 1;
    A[0][0] = 0.f;
    for (int j = 1; j < N; ++j) { A[0][j] = 1.f; A[j][0] = 1.f; }
    for (int i = 1; i < N; ++i)
        for (int j = 1; j < N; ++j)
            A[i][j] = gram[((size_t)b * MH + (i - 1)) * MH + (j - 1)] + ((i == j) ? ANDLAM : 0.f);
    for (int i = 0; i < N; ++i) A[i][N] = (i == 0) ? 1.f : 0.f;

    for (int c = 0; c < N; ++c) {                       // elimination w/ partial pivot
        int piv = c; float best = fabsf(A[c][c]);
        for (int r = c + 1; r < N; ++r) {
            float v = fabsf(A[r][c]);
            if (v > best) { best = v; piv = r; }
        }
        if (piv != c)
            for (int j = c; j <= N; ++j) { float t = A[c][j]; A[c][j] = A[piv][j]; A[piv][j] = t; }
        float inv = 1.f / A[c][c];
        for (int r = c + 1; r < N; ++r) {
            float f = A[r][c] * inv;
            for (int j = c; j <= N; ++j) A[r][j] -= f * A[c][j];
        }
    }
    float sol[6];
    for (int i = N - 1; i >= 0; --i) {
        float s = A[i][N];
        for (int j = i + 1; j < N; ++j) s -= A[i][j] * sol[j];
        sol[i] = s / A[i][i];
    }
    for (int i = 0; i < n; ++i) alpha[b * MH + i] = sol[i + 1];
}

// xk = sum alpha_i F_i (beta=1); F[slot]=ista(xk); G[slot]=F[slot]-xk
// grid = dim3(DD/256, BB), block = 256
__global__ void update_kernel(float* __restrict__ F, float* __restrict__ G,
                              const float* __restrict__ alpha, const float* __restrict__ x0,
                              int n, int slot,
                              const float* __restrict__ rho_p, const float* __restrict__ lam_p) {
    size_t e = (size_t)blockIdx.x * blockDim.x + threadIdx.x;
    size_t b = blockIdx.y;
    float xk = 0.f;
    for (int i = 0; i < n; ++i)
        xk += alpha[b * MH + i] * F[(b * MH + (size_t)i) * DD + e];
    float rho = *rho_p, lam = *lam_p;
    float x = x0[b * DD + e];
    float u = xk - rho * (xk - x);
    float z = shrinkf(u, rho * lam);
    size_t so = (b * MH + (size_t)slot) * DD + e;
    F[so] = z;
    G[so] = z - xk;
}

// out = ista(F[slot])  (final non-stop-gradient step)
__global__ void final_kernel(const float* __restrict__ F, const float* __restrict__ x0,
                             float* __restrict__ out, int slot,
                             const float* __restrict__ rho_p, const float* __restrict__ lam_p) {
    size_t e = (size_t)blockIdx.x * blockDim.x + threadIdx.x;
    size_t b = blockIdx.y;
    float z = F[(b * MH + (size_t)slot) * DD + e];
    float rho = *rho_p, lam = *lam_p;
    float x = x0[b * DD + e];
    float u = z - rho * (z - x);
    out[b * DD + e] = shrinkf(u, rho * lam);
}

extern "C" void kernel_launch(void* const* d_in, const int* in_sizes, int n_in,
                              void* d_out, int out_size, void* d_ws, size_t ws_size,
                              hipStream_t stream) {
    const float* x0    = (const float*)d_in[0];
    const float* rho_p = (const float*)d_in[1];
    const float* lam_p = (const float*)d_in[2];
    float* out = (float*)d_out;

    float* F     = (float*)d_ws;                    // BB*MH*DD floats
    float* G     = F + (size_t)BB * MH * DD;        // BB*MH*DD floats
    float* gram  = G + (size_t)BB * MH * DD;        // BB*25 floats
    float* alpha = gram + BB * MH * MH;             // BB*5 floats

    dim3 gridE((unsigned)(DD / 256), BB);
    dim3 gridG(GBLK, BB);
    int blk = 256;

    // X0 = x0, F0 = f(x0);  X1 = F0, F1 = f(F0)   (store G = F - X)
    fstep_kernel<<<gridE, blk, 0, stream>>>(x0, DD, F, G, 0, x0, rho_p, lam_p);
    fstep_kernel<<<gridE, blk, 0, stream>>>(F, (size_t)MH * DD, F, G, 1, x0, rho_p, lam_p);

    for (int k = 2; k < FTHRES; ++k) {
        int n = (k < MH) ? k : MH;
        int slot = k % MH;
        zero_kernel<<<1, 256, 0, stream>>>(gram, BB * MH * MH);
        gram_kernel<<<gridG, blk, 0, stream>>>(G, gram, n);
        solve_kernel<<<1, 32, 0, stream>>>(gram, alpha, n);
        update_kernel<<<gridE, blk, 0, stream>>>(F, G, alpha, x0, n, slot, rho_p, lam_p);
    }

    final_kernel<<<gridE, blk, 0, stream>>>(F, x0, out, (FTHRES - 1) % MH, rho_p, lam_p);
}